// SE3Norm_2104533975138
// MI455X (gfx1250) — compile-verified
//
#include <hip/hip_runtime.h>
#include <hip/hip_bf16.h>

// SE3Norm for MI455X (gfx1250): memory-bound segment-mean normalize.
// ~320 MB total traffic => ~14us floor @ 23.3 TB/s. No matrix math, so the
// CDNA5-specific paths used are the async global->LDS pipeline (ASYNCcnt,
// double-buffered, s_wait_asynccnt N), wave32 shuffles, b128 vector memory
// ops, and global_prefetch_b8.

#define NB   4096      // number of graphs (fixed by the reference)
#define TILE 256       // nodes staged per LDS tile (256*3 f32 = 3 KB/buffer)
#define EPSV 1e-5f

// ---- CDNA5 async global->LDS copy (per-lane 4B), tracked by ASYNCcnt ------
__device__ __forceinline__ unsigned lds_off_of(const void* p) {
  // Flat LDS addresses carry the wave-relative byte offset in bits [31:0].
  return (unsigned)(uintptr_t)p;
}
__device__ __forceinline__ void async_ld_b32(unsigned lds_byte, const float* g) {
  asm volatile("global_load_async_to_lds_b32 %0, %1, off"
               :: "v"(lds_byte), "v"(g) : "memory");
}
__device__ __forceinline__ void wait_async0() {      // drain everything
  asm volatile("s_wait_asynccnt 0" ::: "memory");
}
__device__ __forceinline__ void wait_async_le3() {   // retire oldest tile,
  asm volatile("s_wait_asynccnt 3" ::: "memory");    // keep next in flight
}

// ---- Kernel 0: segment boundaries via binary search on sorted batch -------
__global__ void __launch_bounds__(256)
bounds_kernel(const int* __restrict__ batch, int n, int* __restrict__ bounds) {
  int idx = blockIdx.x * 256 + threadIdx.x;
  if (idx > NB) return;
  int lo = 0, hi = n;                       // lower_bound(batch, idx)
  while (lo < hi) {
    int mid = (lo + hi) >> 1;
    if (batch[mid] < idx) lo = mid + 1; else hi = mid;
  }
  bounds[idx] = lo;
}

// ---- Kernel 1: per-graph mean norm -> scale[b] = w / (mean + eps) ---------
__global__ void __launch_bounds__(256)
seg_scale_kernel(const float* __restrict__ pos, const int* __restrict__ bounds,
                 const float* __restrict__ weight, float* __restrict__ scale) {
  __shared__ float tile[2][TILE * 3];
  __shared__ float wsum[8];

  const int b     = blockIdx.x;
  const int tid   = threadIdx.x;
  const int start = bounds[b];
  const int end   = bounds[b + 1];
  const int cnt   = end - start;
  const int nfull = cnt / TILE;             // full 256-node tiles
  const int rem   = cnt - nfull * TILE;     // ragged tail nodes

  float acc = 0.0f;

  // -------- pipelined full tiles: exactly 3 async b32 per wave per tile ----
  if (nfull > 0) {
    {  // prologue: tile 0 -> buffer 0
      const float* g = pos + (size_t)3 * (size_t)start;
      async_ld_b32(lds_off_of(&tile[0][tid      ]), g + tid);
      async_ld_b32(lds_off_of(&tile[0][tid + 256]), g + tid + 256);
      async_ld_b32(lds_off_of(&tile[0][tid + 512]), g + tid + 512);
    }
    int cur = 0;
    for (int t = 0; t < nfull; ++t) {
      if (t + 1 < nfull) {                  // stream next tile into other buf
        const float* g = pos + (size_t)3 * (size_t)(start + (t + 1) * TILE);
        float* dst = tile[cur ^ 1];
        async_ld_b32(lds_off_of(&dst[tid      ]), g + tid);
        async_ld_b32(lds_off_of(&dst[tid + 256]), g + tid + 256);
        async_ld_b32(lds_off_of(&dst[tid + 512]), g + tid + 512);
        wait_async_le3();                   // oldest 3 (tile t) retired
      } else {
        wait_async0();                      // last tile: full drain
      }
      __syncthreads();                      // all waves' data visible in LDS
      {
        float x = tile[cur][3 * tid + 0];
        float y = tile[cur][3 * tid + 1];
        float z = tile[cur][3 * tid + 2];
        acc += sqrtf(x * x + y * y + z * z);
      }
      __syncthreads();                      // buffer reusable next iteration
      cur ^= 1;
    }
  }

  // -------- ragged tail tile (unpipelined) ---------------------------------
  if (rem > 0) {
    const int nf = rem * 3;
    const float* g = pos + (size_t)3 * (size_t)(start + nfull * TILE);
    for (int k = tid; k < nf; k += 256)
      async_ld_b32(lds_off_of(&tile[0][k]), g + k);
    wait_async0();
    __syncthreads();
    if (tid < rem) {
      float x = tile[0][3 * tid + 0];
      float y = tile[0][3 * tid + 1];
      float z = tile[0][3 * tid + 2];
      acc += sqrtf(x * x + y * y + z * z);
    }
    __syncthreads();
  }

  // -------- wave32 tree reduction + 8 wave partials (fixed order) ----------
  #pragma unroll
  for (int off = 16; off > 0; off >>= 1) acc += __shfl_down(acc, off);
  if ((tid & 31) == 0) wsum[tid >> 5] = acc;
  __syncthreads();
  if (tid < 32) {
    float v = (tid < 8) ? wsum[tid] : 0.0f;
    #pragma unroll
    for (int off = 4; off > 0; off >>= 1) v += __shfl_down(v, off);
    if (tid == 0) {
      const float mean = v / (float)max(cnt, 1);   // empty segment -> 0
      scale[b] = weight[0] / (mean + EPSV);
    }
  }
}

// ---- Kernel 2: out[i,:] = pos[i,:] * scale[batch[i]]  (4 nodes/thread) ----
__global__ void __launch_bounds__(256)
normalize_kernel(const float* __restrict__ pos, const int* __restrict__ batch,
                 const float* __restrict__ scale, float* __restrict__ out, int n) {
  const int t  = blockIdx.x * 256 + threadIdx.x;   // quad id (4 nodes)
  const int i0 = t * 4;
  if (i0 >= n) return;

  if (i0 + 4 <= n) {                               // full quad: b128 path
    const float4* p4 = (const float4*)pos;
    float4*       o4 = (float4*)out;
    // Speculative stream prefetch ~96 KB ahead (gfx1250 global_prefetch_b8).
    if (i0 + 8192 < n) __builtin_prefetch(p4 + (size_t)3 * (size_t)(t + 2048), 0, 0);

    const int4   bb = ((const int4*)batch)[t];
    const float4 p0 = p4[(size_t)3 * t + 0];       // n0.xyz | n1.x
    const float4 p1 = p4[(size_t)3 * t + 1];       // n1.yz  | n2.xy
    const float4 p2 = p4[(size_t)3 * t + 2];       // n2.z   | n3.xyz
    const float  s0 = scale[bb.x];
    const float  s1 = scale[bb.y];
    const float  s2 = scale[bb.z];
    const float  s3 = scale[bb.w];

    o4[(size_t)3 * t + 0] = make_float4(p0.x * s0, p0.y * s0, p0.z * s0, p0.w * s1);
    o4[(size_t)3 * t + 1] = make_float4(p1.x * s1, p1.y * s1, p1.z * s2, p1.w * s2);
    o4[(size_t)3 * t + 2] = make_float4(p2.x * s2, p2.y * s3, p2.z * s3, p2.w * s3);
  } else {                                         // ragged tail: scalar
    for (int i = i0; i < n; ++i) {
      const float  s = scale[batch[i]];
      const size_t j = (size_t)3 * (size_t)i;
      out[j + 0] = pos[j + 0] * s;
      out[j + 1] = pos[j + 1] * s;
      out[j + 2] = pos[j + 2] * s;
    }
  }
}

extern "C" void kernel_launch(void* const* d_in, const int* in_sizes, int n_in,
                              void* d_out, int out_size, void* d_ws, size_t ws_size,
                              hipStream_t stream) {
  const float* pos    = (const float*)d_in[0];   // [N,3] f32
  const int*   batch  = (const int*)  d_in[1];   // [N]   int (sorted)
  const float* weight = (const float*)d_in[2];   // [1,1] f32
  float*       out    = (float*)d_out;           // [N,3] f32
  const int n = in_sizes[1];                     // N nodes

  // Workspace layout: [0, NB) f32 scale ; then (NB+1) int bounds (~33 KB)
  float* scale  = (float*)d_ws;
  int*   bounds = (int*)((char*)d_ws + (size_t)NB * sizeof(float));

  const int quads = (n + 3) / 4;
  bounds_kernel   <<<(NB + 1 + 255) / 256, 256, 0, stream>>>(batch, n, bounds);
  seg_scale_kernel<<<NB,                   256, 0, stream>>>(pos, bounds, weight, scale);
  normalize_kernel<<<(quads + 255) / 256,  256, 0, stream>>>(pos, batch, scale, out, n);
}